// GIN_32719060861414
// MI455X (gfx1250) — compile-verified
//
#include <hip/hip_runtime.h>
#include <hip/hip_bf16.h>

#define N_NODES   50000
#define N_EDGES   800000
#define DIM       128
#define N_LAYERS  4
#define NUM_GRAPHS 128
#define BN_EPS    1e-5f

typedef __attribute__((ext_vector_type(2))) float v2f;
typedef __attribute__((ext_vector_type(8))) float v8f;

// ---------------------------------------------------------------- zero
__global__ __launch_bounds__(256) void gin_zero(float* __restrict__ p, long n) {
    long i = (long)blockIdx.x * blockDim.x + threadIdx.x;
    if (i < n) p[i] = 0.0f;
}

// ---------------------------------------------------------------- edge scatter-add
// one wave per edge; 32 lanes x float4 = 128 dims
__global__ __launch_bounds__(256) void gin_scatter(const float* __restrict__ hin,
                                                   const int* __restrict__ src,
                                                   const int* __restrict__ dst,
                                                   float* __restrict__ agg) {
    long t = (long)blockIdx.x * blockDim.x + threadIdx.x;
    long e = t >> 5;
    int lane = (int)(t & 31);
    if (e >= N_EDGES) return;
    int s = src[e];
    int d = dst[e];
    const float4* hs = (const float4*)(hin + (long)s * DIM);
    float4 v = hs[lane];
    float* ad = agg + (long)d * DIM + lane * 4;
    atomicAdd(ad + 0, v.x);
    atomicAdd(ad + 1, v.y);
    atomicAdd(ad + 2, v.z);
    atomicAdd(ad + 3, v.w);
}

// ---------------------------------------------------------------- fused GIN MLP via WMMA
// blockDim=256 (8 wave32). Block handles 16 node rows; wave w handles cols [16w,16w+16).
// z = (1+eps)*h + agg ; y1 = relu(z@W1+b1) ; y2 = y1@W2+b2 (+ BN stats atomics)
__global__ __launch_bounds__(256) void gin_mlp(const float* __restrict__ hin,
                                               const float* __restrict__ agg,
                                               const float* __restrict__ W1,
                                               const float* __restrict__ b1,
                                               const float* __restrict__ W2,
                                               const float* __restrict__ b2,
                                               const float* __restrict__ epsp,
                                               float* __restrict__ hout,
                                               float* __restrict__ stats,
                                               int do_stats) {
    __shared__ float zt[16 * DIM];   // A tile for GEMM1
    __shared__ float y1[16 * DIM];   // intermediate (relu output), A tile for GEMM2

    const int row0 = blockIdx.x * 16;
    const int tid  = threadIdx.x;
    const float e1 = 1.0f + epsp[0];

    // cooperative load: z tile (16 x 128)
    for (int i = tid; i < 16 * DIM; i += 256) {
        int r = i >> 7;
        int c = i & (DIM - 1);
        long g = (long)(row0 + r) * DIM + c;
        zt[i] = e1 * hin[g] + agg[g];
    }
    __syncthreads();

    const int wave = tid >> 5;       // 0..7
    const int lane = tid & 31;
    const int n    = lane & 15;      // output column within tile / A row (M)
    const int hi   = lane >> 4;      // lane half: selects K pair (A) / K rows (B), M offset (C/D)
    const int col0 = wave * 16;
    const int kk   = hi * 2;

    __builtin_prefetch(&W1[col0], 0, 0);
    __builtin_prefetch(&W2[col0], 0, 0);

    // ---- GEMM1: acc = z @ W1 + b1 (bias folded into C init; C/D col = n for all 8 regs)
    v8f acc;
    {
        float bv = b1[col0 + n];
        for (int v = 0; v < 8; ++v) acc[v] = bv;
    }
    for (int k = 0; k < DIM; k += 4) {
        v2f a, b;
        const float* zp = &zt[n * DIM + k + kk];   // A: M=n, K=k+kk, k+kk+1
        a.x = zp[0];
        a.y = zp[1];
        b.x = W1[(k + kk) * DIM + col0 + n];       // B: rows K=k+kk / +1, col = col0+n
        b.y = W1[(k + kk + 1) * DIM + col0 + n];
        acc = __builtin_amdgcn_wmma_f32_16x16x4_f32(false, a, false, b,
                                                    (short)0, acc, false, false);
    }
    // relu + stash to LDS: element v -> row m = v + 8*hi, col col0+n
    for (int v = 0; v < 8; ++v) {
        int m = v + 8 * hi;
        y1[m * DIM + col0 + n] = fmaxf(acc[v], 0.0f);
    }
    __syncthreads();

    // ---- GEMM2: acc = y1 @ W2 + b2
    {
        float bv = b2[col0 + n];
        for (int v = 0; v < 8; ++v) acc[v] = bv;
    }
    for (int k = 0; k < DIM; k += 4) {
        v2f a, b;
        const float* yp = &y1[n * DIM + k + kk];
        a.x = yp[0];
        a.y = yp[1];
        b.x = W2[(k + kk) * DIM + col0 + n];
        b.y = W2[(k + kk + 1) * DIM + col0 + n];
        acc = __builtin_amdgcn_wmma_f32_16x16x4_f32(false, a, false, b,
                                                    (short)0, acc, false, false);
    }

    // write out + per-column BN statistics
    float s = 0.0f, ssq = 0.0f;
    for (int v = 0; v < 8; ++v) {
        int m = v + 8 * hi;
        float val = acc[v];
        hout[(long)(row0 + m) * DIM + col0 + n] = val;
        s += val;
        ssq += val * val;
    }
    if (do_stats) {
        atomicAdd(&stats[col0 + n], s);
        atomicAdd(&stats[DIM + col0 + n], ssq);
    }
}

// ---------------------------------------------------------------- BatchNorm (train stats) + ReLU
__global__ __launch_bounds__(256) void gin_bn_relu(float* __restrict__ z,
                                                   const float* __restrict__ stats,
                                                   const float* __restrict__ gamma,
                                                   const float* __restrict__ beta) {
    long i = (long)blockIdx.x * blockDim.x + threadIdx.x;
    if (i >= (long)N_NODES * DIM) return;
    int c = (int)(i & (DIM - 1));
    const float invN = 1.0f / (float)N_NODES;
    float mu  = stats[c] * invN;
    float var = stats[DIM + c] * invN - mu * mu;
    float v = (z[i] - mu) * rsqrtf(var + BN_EPS) * gamma[c] + beta[c];
    z[i] = fmaxf(v, 0.0f);
}

// ---------------------------------------------------------------- global_add_pool
__global__ __launch_bounds__(256) void gin_pool(const float* __restrict__ h,
                                                const int* __restrict__ batch,
                                                float* __restrict__ out) {
    long t = (long)blockIdx.x * blockDim.x + threadIdx.x;
    long nidx = t >> 5;
    int lane = (int)(t & 31);
    if (nidx >= N_NODES) return;
    int g = batch[nidx];
    const float4* hp = (const float4*)(h + nidx * DIM);
    float4 v = hp[lane];
    float* op = out + (long)g * DIM + lane * 4;
    atomicAdd(op + 0, v.x);
    atomicAdd(op + 1, v.y);
    atomicAdd(op + 2, v.z);
    atomicAdd(op + 3, v.w);
}

// ================================================================ host launcher
extern "C" void kernel_launch(void* const* d_in, const int* in_sizes, int n_in,
                              void* d_out, int out_size, void* d_ws, size_t ws_size,
                              hipStream_t stream) {
    const float* x     = (const float*)d_in[0];
    const int*   ei    = (const int*)d_in[1];     // [2, N_EDGES]: src then dst
    const int*   batch = (const int*)d_in[2];
    const float* W1    = (const float*)d_in[3];   // [L, D, D]
    const float* b1    = (const float*)d_in[4];   // [L, D]
    const float* W2    = (const float*)d_in[5];
    const float* b2    = (const float*)d_in[6];
    const float* eps   = (const float*)d_in[7];   // [L]
    const float* gamma = (const float*)d_in[8];   // [L-1, D]
    const float* beta  = (const float*)d_in[9];
    float* out = (float*)d_out;

    const long ND = (long)N_NODES * DIM;
    float* hA    = (float*)d_ws;        // ping
    float* hB    = hA + ND;             // pong
    float* agg   = hB + ND;             // aggregation buffer
    float* stats = agg + ND;            // [2*DIM] sum / sumsq (adjacent to agg)

    const int* src = ei;
    const int* dst = ei + N_EDGES;

    const long zeroN   = ND + 2 * DIM;                  // agg + stats together
    const int  zeroGrd = (int)((zeroN + 255) / 256);
    const int  scatGrd = (int)(((long)N_EDGES * 32 + 255) / 256);
    const int  mlpGrd  = N_NODES / 16;                  // 50000/16 = 3125 exact
    const int  bnGrd   = (int)((ND + 255) / 256);
    const int  poolGrd = (int)(((long)N_NODES * 32 + 255) / 256);

    for (int i = 0; i < N_LAYERS; ++i) {
        const float* hin  = (i == 0) ? x : ((i & 1) ? hA : hB);
        float*       hout = (i & 1) ? hB : hA;
        int do_stats = (i < N_LAYERS - 1) ? 1 : 0;

        gin_zero<<<zeroGrd, 256, 0, stream>>>(agg, zeroN);
        gin_scatter<<<scatGrd, 256, 0, stream>>>(hin, src, dst, agg);
        gin_mlp<<<mlpGrd, 256, 0, stream>>>(hin, agg,
                                            W1 + (long)i * DIM * DIM, b1 + (long)i * DIM,
                                            W2 + (long)i * DIM * DIM, b2 + (long)i * DIM,
                                            eps + i, hout, stats, do_stats);
        if (do_stats) {
            gin_bn_relu<<<bnGrd, 256, 0, stream>>>(hout, stats,
                                                   gamma + (long)i * DIM,
                                                   beta + (long)i * DIM);
        }
    }
    // final h lives in hB (layer 3 output)
    gin_zero<<<(NUM_GRAPHS * DIM + 255) / 256, 256, 0, stream>>>(out, NUM_GRAPHS * DIM);
    gin_pool<<<poolGrd, 256, 0, stream>>>(hB, batch, out);
}